// MoERouter_592705487374
// MI455X (gfx1250) — compile-verified
//
#include <hip/hip_runtime.h>
#include <math.h>

// MoE router for MI455X (gfx1250, wave32, WMMA).
// logits = x[T,H] @ W[E,H]^T + bias; softmax; top-8; renormalize.
// T=16384, H=2048, E=64. GEMM via V_WMMA_F32_16X16X4_F32 (full f32 precision).
//
// HBM-bound: 128 MiB of x streamed once (floor ~5.8 us @ 23.3 TB/s).
// - K split across 4 waves per 16-token tile (partials reduced via ds_add_f32)
//   -> 4096 waves (~4/SIMD) for latency hiding.
// - Explicit depth-2 software pipeline so loads for chunk k+8 are in flight
//   while WMMAs consume chunk k (avoids s_wait_loadcnt 0 serialization).
// - H templated so B loads share one 32-bit offset VGPR + immediate offsets.

typedef __attribute__((ext_vector_type(2))) float v2f;
typedef __attribute__((ext_vector_type(8))) float v8f;

#define EXPERTS        64
#define TOPK_N         8
#define KSLICES        4     // waves cooperating on one token tile (K-split)
#define TILES_PER_BLK  2     // token tiles per block
#define WAVES          (KSLICES * TILES_PER_BLK)   // 8 waves = 256 threads
#define ROWS_PER_TILE  16
#define LDS_STRIDE     65    // 64 + 1 pad -> conflict-free row-serial reads

template <int KH>            // KH = compile-time H (0 -> runtime H)
__global__ __launch_bounds__(WAVES * 32)
void moe_router_kernel(const float* __restrict__ x,
                       const float* __restrict__ bias,
                       const float* __restrict__ W,
                       float* __restrict__ outW,
                       float* __restrict__ outI,
                       int T, int Hrt)
{
    const int H = KH ? KH : Hrt;

    __shared__ float lds[TILES_PER_BLK * ROWS_PER_TILE * LDS_STRIDE];

    const int tid    = threadIdx.x;
    const int lane   = tid & 31;
    const int wave   = tid >> 5;
    const int kslice = wave & (KSLICES - 1);
    const int tsub   = wave >> 2;                 // token tile within block
    const int l      = lane & 15;                 // M (A) / N (B,C) index in tile
    const int half   = lane >> 4;                 // K sub-slot select
    const int tile   = blockIdx.x * TILES_PER_BLK + tsub;
    const int r0     = tile * ROWS_PER_TILE;
    const bool valid = (r0 < T);                  // wave-uniform

    // ---- zero the reduction buffer ----
    for (int i = tid; i < TILES_PER_BLK * ROWS_PER_TILE * LDS_STRIDE; i += WAVES * 32)
        lds[i] = 0.0f;
    __syncthreads();

    if (valid) {
        const int kLen = H / KSLICES;             // 512 for H=2048
        const int k0   = kslice * kLen;

        // 32-bit element offsets from uniform bases (saddr + voffset form).
        // A: lane<16 -> (M=l, K=k..k+1), lane>=16 -> (M=l, K=k+2..k+3)
        const unsigned offA = (unsigned)(r0 + l) * (unsigned)H + k0 + 2 * half;
        // B: B[k][n] = W[n][k]; all 4 expert tiles share one offset VGPR,
        //    per-tile delta n*16*H is an immediate when KH != 0.
        const unsigned offB = (unsigned)l * (unsigned)H + k0 + 2 * half;

        v8f acc0 = {}; v8f acc1 = {}; v8f acc2 = {}; v8f acc3 = {};

        v2f aC, b0C, b1C, b2C, b3C;   // chunk being consumed
        v2f aN, b0N, b1N, b2N, b3N;   // chunk in flight

#define LD_CHUNK(A_, B0_, B1_, B2_, B3_, kk)                                   \
        do {                                                                   \
            A_  = *(const v2f*)(x + offA + (kk));                              \
            B0_ = *(const v2f*)(W + offB + 0u * 16u * H + (kk));               \
            B1_ = *(const v2f*)(W + offB + 1u * 16u * H + (kk));               \
            B2_ = *(const v2f*)(W + offB + 2u * 16u * H + (kk));               \
            B3_ = *(const v2f*)(W + offB + 3u * 16u * H + (kk));               \
        } while (0)

#define DO_WMMA()                                                              \
        do {                                                                   \
            acc0 = __builtin_amdgcn_wmma_f32_16x16x4_f32(false, aC, false, b0C,\
                                                  (short)0, acc0, false, false);\
            acc1 = __builtin_amdgcn_wmma_f32_16x16x4_f32(false, aC, false, b1C,\
                                                  (short)0, acc1, false, false);\
            acc2 = __builtin_amdgcn_wmma_f32_16x16x4_f32(false, aC, false, b2C,\
                                                  (short)0, acc2, false, false);\
            acc3 = __builtin_amdgcn_wmma_f32_16x16x4_f32(false, aC, false, b3C,\
                                                  (short)0, acc3, false, false);\
        } while (0)

#define ROTATE()                                                               \
        do { aC = aN; b0C = b0N; b1C = b1N; b2C = b2N; b3C = b3N; } while (0)

        LD_CHUNK(aC, b0C, b1C, b2C, b3C, 0);
        LD_CHUNK(aN, b0N, b1N, b2N, b3N, 4);

        #pragma unroll 2
        for (int k = 0; k < kLen - 8; k += 4) {
            v2f aF, b0F, b1F, b2F, b3F;
            LD_CHUNK(aF, b0F, b1F, b2F, b3F, k + 8);   // keep ~10 loads in flight
            DO_WMMA();                                  // consumes chunk k
            ROTATE();
            aN = aF; b0N = b0F; b1N = b1F; b2N = b2F; b3N = b3F;
        }
        DO_WMMA();   // chunk kLen-8
        ROTATE();
        DO_WMMA();   // chunk kLen-4

#undef LD_CHUNK
#undef DO_WMMA
#undef ROTATE

        // ---- reduce partial C tiles across the 4 k-slice waves ----
        float* rowbase = lds + tsub * ROWS_PER_TILE * LDS_STRIDE;
        #pragma unroll
        for (int j = 0; j < 8; ++j) {
            const int m = j + 8 * half;           // C layout: VGPR j -> M=j / M=j+8
            atomicAdd(&rowbase[m * LDS_STRIDE + 0 * 16 + l], acc0[j]);
            atomicAdd(&rowbase[m * LDS_STRIDE + 1 * 16 + l], acc1[j]);
            atomicAdd(&rowbase[m * LDS_STRIDE + 2 * 16 + l], acc2[j]);
            atomicAdd(&rowbase[m * LDS_STRIDE + 3 * 16 + l], acc3[j]);
        }
    }

    __syncthreads();

    // ---- epilogue: one lane per token row; first wave handles all 32 rows ----
    if (tid < TILES_PER_BLK * ROWS_PER_TILE) {
        const int tsub2 = tid >> 4;
        const int r     = tid & 15;
        const int token = (blockIdx.x * TILES_PER_BLK + tsub2) * ROWS_PER_TILE + r;
        if (token < T) {
            float* row = lds + (tsub2 * ROWS_PER_TILE + r) * LDS_STRIDE;

            // add bias, find max
            float m = -INFINITY;
            #pragma unroll 4
            for (int e = 0; e < EXPERTS; ++e) {
                const float v = row[e] + bias[e];
                row[e] = v;
                m = fmaxf(m, v);
            }

            // full softmax denominator (reference divides by global Z)
            float Z = 0.0f;
            #pragma unroll 4
            for (int e = 0; e < EXPERTS; ++e) Z += expf(row[e] - m);

            float p[TOPK_N];
            int   idx[TOPK_N];
            float s = 0.0f;
            #pragma unroll
            for (int i = 0; i < TOPK_N; ++i) {
                float best = -INFINITY;
                int   bi   = 0;
                for (int e = 0; e < EXPERTS; ++e) {
                    const float v = row[e];
                    if (v > best) { best = v; bi = e; }  // ties -> lowest index
                }
                row[bi] = -INFINITY;                     // mask for next pass
                const float pi = expf(best - m) / Z;
                p[i] = pi; idx[i] = bi; s += pi;
            }

            const float inv = 1.0f / (s + 1e-9f);
            #pragma unroll
            for (int i = 0; i < TOPK_N; ++i) {
                outW[(size_t)token * TOPK_N + i] = p[i] * inv;
                outI[(size_t)token * TOPK_N + i] = (float)idx[i]; // exact in f32
            }
        }
    }
}

extern "C" void kernel_launch(void* const* d_in, const int* in_sizes, int n_in,
                              void* d_out, int out_size, void* d_ws, size_t ws_size,
                              hipStream_t stream)
{
    const float* x    = (const float*)d_in[0];
    const float* bias = (const float*)d_in[1];
    const float* W    = (const float*)d_in[2];

    const int E = in_sizes[1];            // 64
    const int H = in_sizes[2] / E;        // 2048
    const int T = in_sizes[0] / H;        // 16384 tokens
    (void)n_in; (void)out_size; (void)d_ws; (void)ws_size;

    float* outW = (float*)d_out;                       // [T, 8] weights
    float* outI = (float*)d_out + (size_t)T * TOPK_N;  // [T, 8] indices (as f32 values)

    const int tokensPerBlock = TILES_PER_BLK * ROWS_PER_TILE;  // 32
    const int grid = (T + tokensPerBlock - 1) / tokensPerBlock;

    if (H == 2048) {
        moe_router_kernel<2048><<<grid, WAVES * 32, 0, stream>>>(x, bias, W,
                                                                 outW, outI, T, H);
    } else {
        moe_router_kernel<0><<<grid, WAVES * 32, 0, stream>>>(x, bias, W,
                                                              outW, outI, T, H);
    }
}